// kernel_generated_1_62904091017235
// MI455X (gfx1250) — compile-verified
//
#include <hip/hip_runtime.h>

// CDNA5 / gfx1250 fused kernel for:
//   t4 = einsum('kjmnl,ji->kmnli', x, w2)          (channel mix, K=64 -> P=8)
//   double unfold over width l with nested zero-pad masks
//   t6 = einsum('mlonikp,likj->monpj', t5, w1)     (M=28, N=32, K=36 windowed mix)
//   roll(+1) over o, output (128, 256, 28, 28) fp32
//
// HBM-bound (~205 MB traffic vs 2.26 GFLOP): single fused pass. x staged to
// LDS with async global->LDS (ASYNCcnt), both GEMMs on V_WMMA_F32_16X16X4_F32.
// Weight panels are stored fragment-major so every B fragment is a single
// ds_load_b64 at an immediate offset; stage-2 A offsets/masks are constexpr-
// folded with a single cndmask on the lane's half bit. Coalesced stores.

typedef float v2f __attribute__((ext_vector_type(2)));
typedef float v8f __attribute__((ext_vector_type(8)));

#define JCH 64   // input channels (contracted by w2)
#define LC  4    // m dimension of x
#define OO  28   // first spatial dim (no conv, only roll)
#define NN  28   // second spatial dim (double-unfolded)
#define PP  8    // w2 output channels
#define JJ  32   // w1 output channels
#define T4STR 36 // padded row stride of t4pad (2-halo each side + slack)

// stage-2 A-offset: kk=(lc*9+bb*3+aa); element index = lc*288 + p*36 + n+aa+bb
__host__ __device__ constexpr int offc(int kk) {
    return (kk / 9) * 288 + ((kk % 9) / 3) + ((kk % 9) % 3);
}
// residual nested-unfold masks (everything else is handled by halo zeros)
__host__ __device__ constexpr bool m2(int kk) { return (kk % 9) == 2; } // zero @ n==0
__host__ __device__ constexpr bool m6(int kk) { return (kk % 9) == 6; } // zero @ n==27

__launch_bounds__(256, 2)
__global__ void fused_mix_unfold_mix(const float* __restrict__ x,
                                     const float* __restrict__ w1,
                                     const float* __restrict__ w2,
                                     float* __restrict__ out)
{
    // buf: stage A = x slab [64*4][28] (7168 f); stage C = t6 [8][28][33] (7392 f)
    __shared__ __align__(16) float buf[7392];
    __shared__ float t4pad[LC * PP * T4STR];  // [lc*8+p][36], halo-zeroed = 1152 f
    __shared__ float w1f[1152];  // frag-major: [g=2s+kbi][j][2], val=w1[(2g+d)*32+j]
    __shared__ float w2f[1024];  // frag-major: [g=2ks+kbi][c][2], val=w2pad[2g+d][c]

    const int tid   = threadIdx.x;
    const int lane  = tid & 31;
    const int wave  = tid >> 5;     // 8 waves
    const int half  = lane >> 4;    // 0 = lanes 0-15, 1 = lanes 16-31
    const int l16   = lane & 15;
    const int batch = blockIdx.x / OO;
    const int o     = blockIdx.x % OO;

    // ---------------- async-stage x slab into LDS, repack weights ------------
    // x[batch, j2, lc, o, n]: row = j2*4+lc has stride 784 floats; take 28 floats.
    {
        const float4* xb4 = (const float4*)(x + (size_t)batch * (JCH * LC * OO * NN) + (size_t)o * NN);
        float4* buf4 = (float4*)buf;
        int row = tid / 7, q = tid - (tid / 7) * 7;      // chunk = row*7+q
        #pragma unroll
        for (int k = 0; k < 7; ++k) {                    // 1792 x 16B chunks
            const unsigned long long ga = (unsigned long long)(xb4 + row * 196 + q);
            const unsigned int       la = (unsigned int)(unsigned long long)(buf4 + row * 7 + q);
            asm volatile("global_load_async_to_lds_b128 %0, %1, off"
                         :: "v"(la), "v"(ga) : "memory");
            row += 36; q += 4;                           // +256 chunks, div-free
            if (q >= 7) { q -= 7; row += 1; }
        }
        // w1f[g*64 + j*2 + d] = w1[(2g+d)*32 + j],  g=2s+kbi
        for (int idx = tid; idx < 1152; idx += 256) {
            const int g = idx >> 6, j = (idx >> 1) & 31, d = idx & 1;
            w1f[idx] = w1[(2 * g + d) * JJ + j];
        }
        // w2f[g*32 + c*2 + d] = (c<8) ? w2[(2g+d)*8 + c] : 0,  g=2ks+kbi
        for (int idx = tid; idx < 1024; idx += 256) {
            const int g = idx >> 5, c = (idx >> 1) & 15, d = idx & 1;
            w2f[idx] = (c < PP) ? w2[(2 * g + d) * PP + c] : 0.0f;
        }
        for (int idx = tid; idx < LC * PP * T4STR; idx += 256) t4pad[idx] = 0.0f;
        asm volatile("s_wait_asynccnt 0x0" ::: "memory");
    }
    __syncthreads();

    // ---------------- stage 1: t4[r=(lc*28+n)][p] = sum_j2 xs[j2][r]*w2[j2][p]
    // M = 112 rows (7 tiles of 16), N = 8 (padded to 16), K = 64 (16 steps of 4).
    // Wave w < 7 owns M-tile w.  A layout: lane=M, VGPR0 K={0,2}, VGPR1 K={1,3}.
    if (wave < 7) {
        v8f acc = {};
        const int r = wave * 16 + l16;                    // A-matrix row (M)
        const float* pA = &buf[half * 224 + r];           // + 448*ks, +112 for .y
        const float* pB = &w2f[half * 32 + l16 * 2];      // + 64*ks (b64 pair)
        #pragma unroll
        for (int ks = 0; ks < 16; ++ks) {
            v2f a, b;
            a.x = pA[448 * ks];
            a.y = pA[448 * ks + 112];
            b   = *(const v2f*)(pB + 64 * ks);            // (w2[k0][p], w2[k0+1][p])
            acc = __builtin_amdgcn_wmma_f32_16x16x4_f32(
                      false, a, false, b, (short)0, acc, false, false);
        }
        // C layout: VGPR v -> row M = v + half*8, col N = l16 (= p, keep p<8)
        if (l16 < PP) {
            #pragma unroll
            for (int v = 0; v < 8; ++v) {
                const int row = wave * 16 + v + half * 8;  // lc*28+n, < 112
                const int lc  = row / NN;
                const int n   = row - lc * NN;
                t4pad[(lc * PP + l16) * T4STR + 2 + n] = acc[v];
            }
        }
    }
    __syncthreads();

    // ---------------- stage 2: per wave p = wave:
    // out[n, j] = sum_{kk<36} A2[n,kk] * w1[kk][j]
    // A2[n, kk] = t4[lc][p][n+aa+bb-2]; halo zeros cover the l2 bound, the
    // first-unfold bound survives in only 8 constexpr-selected lane cases.
    {
        const int p = wave;
        const float* pA0 = &t4pad[p * T4STR + l16];       // mt0 rows (n = l16)
        const float* pB  = &w1f[half * 64 + l16 * 2];     // + 128*s (+32 for nt1)
        const bool hiPad = (l16 >= 12);                   // mt1 rows n>=28 -> 0

        v8f acc00 = {}, acc01 = {}, acc10 = {}, acc11 = {};
        #pragma unroll
        for (int s = 0; s < 9; ++s) {                     // K = 36 = 9 steps of 4
            const int offx = half ? offc(4 * s + 2) : offc(4 * s + 0);
            const int offy = half ? offc(4 * s + 3) : offc(4 * s + 1);
            float a0x = pA0[offx];
            float a0y = pA0[offy];
            float a1x = pA0[offx + 16];                   // mt1: n = 16 + l16
            float a1y = pA0[offy + 16];
            const bool z2x = half ? m2(4 * s + 2) : m2(4 * s + 0);
            const bool z2y = half ? m2(4 * s + 3) : m2(4 * s + 1);
            const bool z6x = half ? m6(4 * s + 2) : m6(4 * s + 0);
            const bool z6y = half ? m6(4 * s + 3) : m6(4 * s + 1);
            a0x = (z2x && l16 == 0) ? 0.0f : a0x;
            a0y = (z2y && l16 == 0) ? 0.0f : a0y;
            a1x = (hiPad || (z6x && l16 == 11)) ? 0.0f : a1x;
            a1y = (hiPad || (z6y && l16 == 11)) ? 0.0f : a1y;

            const v2f a0 = {a0x, a0y};
            const v2f a1 = {a1x, a1y};
            const v2f b0 = *(const v2f*)(pB + 128 * s);        // j = l16
            const v2f b1 = *(const v2f*)(pB + 128 * s + 32);   // j = 16 + l16
            acc00 = __builtin_amdgcn_wmma_f32_16x16x4_f32(false, a0, false, b0, (short)0, acc00, false, false);
            acc01 = __builtin_amdgcn_wmma_f32_16x16x4_f32(false, a0, false, b1, (short)0, acc01, false, false);
            acc10 = __builtin_amdgcn_wmma_f32_16x16x4_f32(false, a1, false, b0, (short)0, acc10, false, false);
            acc11 = __builtin_amdgcn_wmma_f32_16x16x4_f32(false, a1, false, b1, (short)0, acc11, false, false);
        }

        // store to padded LDS tile t6s[p][n][j] (stride 33 -> no bank conflicts)
        #pragma unroll
        for (int v = 0; v < 8; ++v) {
            const int n0 = v + half * 8;        // M-tile 0 row, < 16
            const int n1 = 16 + n0;             // M-tile 1 row, < 32
            buf[(p * NN + n0) * 33 + l16]      = acc00[v];
            buf[(p * NN + n0) * 33 + 16 + l16] = acc01[v];
            if (n1 < NN) {
                buf[(p * NN + n1) * 33 + l16]      = acc10[v];
                buf[(p * NN + n1) * 33 + 16 + l16] = acc11[v];
            }
        }
    }
    __syncthreads();

    // ---------------- coalesced store with roll folded in ---------------------
    // out[batch, c=j*8+p, o_out=(o+1)%28, n]; element idx = c*28+n = tid+256k.
    {
        const int oout = (o + 1) % OO;
        float* ob = out + (size_t)batch * (256 * OO * NN) + (size_t)oout * NN;
        int c = tid / NN, n = tid - (tid / NN) * NN;
        #pragma unroll
        for (int k = 0; k < 28; ++k) {
            ob[c * (OO * NN) + n] = buf[((c & 7) * NN + n) * 33 + (c >> 3)];
            c += 9; n += 4;                                 // +256 elements, div-free
            if (n >= NN) { n -= NN; c += 1; }
        }
    }
}

extern "C" void kernel_launch(void* const* d_in, const int* in_sizes, int n_in,
                              void* d_out, int out_size, void* d_ws, size_t ws_size,
                              hipStream_t stream) {
    const float* x  = (const float*)d_in[0];   // (128, 64, 4, 28, 28)
    const float* w1 = (const float*)d_in[1];   // (4, 3, 3, 32)
    const float* w2 = (const float*)d_in[2];   // (64, 8)
    float* out = (float*)d_out;                // (128, 256, 28, 28)

    dim3 grid(128 * 28);
    dim3 block(256);
    fused_mix_unfold_mix<<<grid, block, 0, stream>>>(x, w1, w2, out);
}